// PatchDISTANCELoss_24962349924824
// MI455X (gfx1250) — compile-verified
//
#include <hip/hip_runtime.h>

typedef __attribute__((ext_vector_type(2))) float v2f;
typedef __attribute__((ext_vector_type(4))) float v4f;
typedef __attribute__((ext_vector_type(8))) float v8f;

#define N_PATCH 2048
#define DIM 256
#define WPB 8  // waves per block (256 threads, wave32)

__global__ __launch_bounds__(256) void patch_l1_wmma(const float* __restrict__ fq,
                                                     const float* __restrict__ fk,
                                                     float* __restrict__ partial) {
    __shared__ float shq[WPB][DIM];
    __shared__ float shk[WPB][DIM];
    __shared__ float swave[WPB];

    const int tid  = threadIdx.x;
    const int wave = tid >> 5;
    const int lane = tid & 31;
    const int patch = blockIdx.x * WPB + wave;

    // Stage this wave's patch rows (2 x 1KB) into LDS with b128 loads.
    {
        const v4f* qv = (const v4f*)(fq + (size_t)patch * DIM);
        const v4f* kv = (const v4f*)(fk + (size_t)patch * DIM);
        v4f* sq = (v4f*)shq[wave];
        v4f* sk = (v4f*)shk[wave];
        sq[lane]      = qv[lane];
        sq[lane + 32] = qv[lane + 32];
        sk[lane]      = kv[lane];
        sk[lane + 32] = kv[lane + 32];
    }
    // Intra-wave LDS dependency only; compiler inserts s_wait_dscnt.

    const int  m  = lane & 15;
    const bool lo = lane < 16;

    // Pre-build all 16 B fragments (4x16 f32): row0=q, row1=k, rows2,3=0.
    // Lanes 16-31 (K=2,3 halves) forced to zero.
    v2f B[16];
#pragma unroll
    for (int t = 0; t < 16; ++t) {
        float qq = shq[wave][t * 16 + m];
        float kk = shk[wave][t * 16 + m];
        B[t].x = lo ? qq : 0.0f;
        B[t].y = lo ? kk : 0.0f;
    }

    v8f acc = {0.f, 0.f, 0.f, 0.f, 0.f, 0.f, 0.f, 0.f};

    for (int ti = 0; ti < 16; ++ti) {
        // A fragment (16x4 f32): col0=q_i, col1=-k_i, cols2,3=0.
        float aq = shq[wave][ti * 16 + m];
        float ak = shk[wave][ti * 16 + m];
        v2f A;
        A.x = lo ?  aq : 0.0f;
        A.y = lo ? -ak : 0.0f;

#pragma unroll
        for (int tj = 0; tj < 16; ++tj) {
            v8f c = {0.f, 0.f, 0.f, 0.f, 0.f, 0.f, 0.f, 0.f};
            // D[i][j] = q_i*q_j + (-k_i)*k_j  == the L1 difference tile
            v8f d = __builtin_amdgcn_wmma_f32_16x16x4_f32(
                false, A, false, B[tj], (short)0, c, false, false);
            acc[0] += __builtin_fabsf(d[0]);
            acc[1] += __builtin_fabsf(d[1]);
            acc[2] += __builtin_fabsf(d[2]);
            acc[3] += __builtin_fabsf(d[3]);
            acc[4] += __builtin_fabsf(d[4]);
            acc[5] += __builtin_fabsf(d[5]);
            acc[6] += __builtin_fabsf(d[6]);
            acc[7] += __builtin_fabsf(d[7]);
        }
    }

    // Per-lane pairwise sum of the 8 accumulators.
    float s01 = acc[0] + acc[1];
    float s23 = acc[2] + acc[3];
    float s45 = acc[4] + acc[5];
    float s67 = acc[6] + acc[7];
    float s = (s01 + s23) + (s45 + s67);

    // Wave32 butterfly reduction.
#pragma unroll
    for (int off = 16; off > 0; off >>= 1)
        s += __shfl_xor(s, off, 32);

    if (lane == 0) swave[wave] = s;
    __syncthreads();
    if (tid == 0) {
        float t = 0.0f;
#pragma unroll
        for (int w = 0; w < WPB; ++w) t += swave[w];
        partial[blockIdx.x] = t;
    }
}

__global__ __launch_bounds__(256) void reduce_partials(const float* __restrict__ partial,
                                                       float* __restrict__ out) {
    __shared__ float sh[256];
    const int tid = threadIdx.x;
    sh[tid] = partial[tid];
    __syncthreads();
#pragma unroll
    for (int s = 128; s > 0; s >>= 1) {
        if (tid < s) sh[tid] += sh[tid + s];
        __syncthreads();
    }
    if (tid == 0) {
        // mean over P*D*D = 2048*256*256 = 2^27 elements
        out[0] = sh[0] * (1.0f / 134217728.0f);
    }
}

extern "C" void kernel_launch(void* const* d_in, const int* in_sizes, int n_in,
                              void* d_out, int out_size, void* d_ws, size_t ws_size,
                              hipStream_t stream) {
    const float* fq = (const float*)d_in[0];
    const float* fk = (const float*)d_in[1];
    float* partial = (float*)d_ws;  // 256 floats of scratch

    patch_l1_wmma<<<N_PATCH / WPB, 256, 0, stream>>>(fq, fk, partial);
    reduce_partials<<<1, 256, 0, stream>>>(partial, (float*)d_out);
}